// CstrFlashModel_78297253806377
// MI455X (gfx1250) — compile-verified
//
#include <hip/hip_runtime.h>
#include <math.h>

// ---------------------------------------------------------------------------
// CSTR+flash neural-ODE RK4 rollout, fp32, WMMA_F32_16x16x4_F32 on gfx1250.
// One workgroup = 16 batch rows, 8 wave32s. Whole T=128 scan inside kernel.
// All MLP weights resident in LDS (pair-swizzled B-fragment layout).
// Branch-free tanh via v_exp_f32/v_rcp_f32 (replaces libm tanhf expansion).
// ---------------------------------------------------------------------------

typedef __attribute__((ext_vector_type(2))) float v2f;
typedef __attribute__((ext_vector_type(8))) float v8f;

#define B_TOT   2048
#define T_STEPS 128
#define NXc     8
#define NUc     2
#define NZc     50          // NP*(NY+NU)
#define DINP    64          // 60 padded to 64
#define DH      128
#define BT      16          // batch tile per block
#define DELTA_F 0.01f

// LDS layout (bytes). All offsets 8B-aligned for v2f.
#define LDW     129         // pair-row stride for W1p/W2p (padded)
#define LDW3    17          // pair-row stride for W3p (N padded 8->16, +1)
#define OFF_W1P 0u                                  // v2f[32*129]  = 33024 B
#define OFF_W2P (OFF_W1P + 32u*LDW*8u)              // v2f[64*129]  = 66048 B
#define OFF_W3P (OFF_W2P + 64u*LDW*8u)              // v2f[64*17]   =  8704 B
#define OFF_INP (OFF_W3P + 64u*LDW3*8u)             // f32[16*64]   =  4096 B
#define OFF_H1  (OFF_INP + 16u*64u*4u)              // f32[16*128]  =  8192 B
#define OFF_H2  (OFF_H1  + 16u*128u*4u)             // f32[16*128]  =  8192 B
#define OFF_NN  (OFF_H2  + 16u*128u*4u)             // f32[16*16]   =  1024 B
#define OFF_XG  (OFF_NN  + 256u*4u)                 // f32[16*8]
#define OFF_XS  (OFF_XG  + 128u*4u)                 // f32[16*8]
#define OFF_Z   (OFF_XS  + 128u*4u)                 // f32[16*52]   (stride 52)
#define OFF_K   (OFF_Z   + 16u*52u*4u)              // f32[4*16*8]
#define OFF_U   (OFF_K   + 512u*4u)                 // f32[16*2]
#define OFF_B1  (OFF_U   + 32u*4u)                  // f32[128]
#define OFF_B2  (OFF_B1  + 128u*4u)                 // f32[128]
#define OFF_B3  (OFF_B2  + 128u*4u)                 // f32[8] (pad 16)
#define SMEM_BYTES (OFF_B3 + 16u*4u)

// Branch-free tanh: tanh(x) = 1 - 2/(exp(2x)+1).
// exp(2x)->inf  => 1 ; exp(2x)->0 => -1. No inf/inf, no exec-mask branches.
__device__ __forceinline__ float fast_tanh(float x) {
  float e = __expf(2.0f * x);                 // v_exp_f32 path
  return 1.0f - __fdividef(2.0f, e + 1.0f);   // v_rcp_f32 + mul
}

// 16x16 output tile, A[16xK] row-major f32 in LDS, B pre-swizzled as
// (K/2) x N float2 pairs: Bp[kk][n] = {B[2kk][n], B[2kk+1][n]}.
// A frag: lane holds M=lane&15, K = 4*kk + 2*(lane>>4) + {0,1}.
// B frag: lane holds N=lane&15, same K pairing -> one v2f load each.
__device__ __forceinline__ v8f mm16(const float* __restrict__ A, int lda,
                                    const v2f* __restrict__ Bp, int ldb,
                                    int kIters, int lane) {
  v8f acc = {};
  const int mr = lane & 15;
  const int kh = lane >> 4;            // 0 or 1
  const float* ap = A + mr * lda + 2 * kh;
  const v2f*   bp = Bp + (size_t)kh * ldb + mr;
#pragma unroll
  for (int kk = 0; kk < kIters; ++kk) {
    v2f a = *(const v2f*)(ap + 4 * kk);
    v2f b = bp[(size_t)(2 * kk) * ldb];
    acc = __builtin_amdgcn_wmma_f32_16x16x4_f32(false, a, false, b,
                                                (short)0, acc, false, false);
  }
  return acc;
}

// MLP (layer1+tanh, layer2+tanh, split-K layer3) + physics RHS -> sK[stage].
__device__ __forceinline__ void mlp_and_physics(
    int tid, int wave, int lane, int stage,
    const float* sInp, float* sH1, float* sH2, float* sNN,
    const v2f* W1p, const v2f* W2p, const v2f* W3p,
    const float* sB1, const float* sB2, const float* sB3,
    const float* xsrc, const float* sU, float* sK) {
  // ---- layer 1: [16x64] @ [64x128] ----
  {
    v8f d = mm16(sInp, DINP, W1p + wave * 16, LDW, 16, lane);
    const int nc = wave * 16 + (lane & 15);
    const int kh = lane >> 4;
    const float bias = sB1[nc];
#pragma unroll
    for (int v = 0; v < 8; ++v)
      sH1[(v + 8 * kh) * DH + nc] = fast_tanh(d[v] + bias);
  }
  __syncthreads();
  // ---- layer 2: [16x128] @ [128x128] ----
  {
    v8f d = mm16(sH1, DH, W2p + wave * 16, LDW, 32, lane);
    const int nc = wave * 16 + (lane & 15);
    const int kh = lane >> 4;
    const float bias = sB2[nc];
#pragma unroll
    for (int v = 0; v < 8; ++v)
      sH2[(v + 8 * kh) * DH + nc] = fast_tanh(d[v] + bias);
  }
  sNN[tid] = 0.0f;  // zero layer-3 accumulator (256 == 16*16)
  __syncthreads();
  // ---- layer 3: [16x128] @ [128x16(pad)] split-K across 8 waves ----
  {
    v8f d = mm16(sH2 + wave * 16, DH, W3p + wave * 8 * LDW3, LDW3, 4, lane);
    const int nc = lane & 15;
    const int kh = lane >> 4;
    if (nc < NXc) {
#pragma unroll
      for (int v = 0; v < 8; ++v)
        atomicAdd(&sNN[(v + 8 * kh) * 16 + nc], d[v]);
    }
  }
  __syncthreads();
  // ---- physics RHS + k = fg + nn, one lane per batch row ----
  if (tid < BT) {
    const float* xs = xsrc + tid * NXc;
    float Hr  = xs[0] * 2.0f + 50.0f;
    float CAr = xs[1] * 0.2f + 1.0f;
    float CBr = xs[2] * 0.1f + 0.5f;
    float Tr  = xs[3] * 4.0f + 313.0f;
    float Hb  = xs[4] * 2.0f + 50.0f;
    float CAb = xs[5] * 0.2f + 1.0f;
    float CBb = xs[6] * 0.1f + 0.5f;
    float Tb  = xs[7] * 4.0f + 313.0f;
    float F = sU[tid * 2 + 0] * 1.0f + 10.0f;
    float D = sU[tid * 2 + 1] * 0.5f + 5.0f;
    float Fr = 2.5f * __fsqrt_rn(Hr);
    float Fb = 1.5f * __fsqrt_rn(Hb);
    float invArHr = __fdividef(1.0f, 3.0f * Hr);
    float invAbHb = __fdividef(1.0f, 3.0f * Hb);
    float dHr  = (F + D - Fr) * (1.0f / 3.0f);
    float dCAr = (F * (6.0f - CAr) - D * CAr) * invArHr;
    float dCBr = (-F * CBr - D * CBr) * invArHr;
    float dTr  = (F * (320.0f - Tr) + D * (300.0f - Tr)) * invArHr
               + (200.0f / 30.0f) * invArHr;          // QR/(PHO*CP) * 1/(AR*Hr)
    float dHb  = (Fr - Fb - D) * (1.0f / 3.0f);
    float dCAb = (Fr * (CAr - CAb) + D * CAb) * invAbHb;
    float dCBb = (Fr * (CBr - CBb) + D * CBb) * invAbHb;
    float dTb  = Fr * (Tr - Tb) * invAbHb + (200.0f / 30.0f) * invAbHb;
    float*       kr = sK + stage * (BT * NXc) + tid * NXc;
    const float* nn = sNN + tid * 16;
    kr[0] = dHr  * 0.5f  + nn[0] + sB3[0];
    kr[1] = dCAr * 5.0f  + nn[1] + sB3[1];
    kr[2] = dCBr * 10.0f + nn[2] + sB3[2];
    kr[3] = dTr  * 0.25f + nn[3] + sB3[3];
    kr[4] = dHb  * 0.5f  + nn[4] + sB3[4];
    kr[5] = dCAb * 5.0f  + nn[5] + sB3[5];
    kr[6] = dCBb * 10.0f + nn[6] + sB3[6];
    kr[7] = dTb  * 0.25f + nn[7] + sB3[7];
  }
  __syncthreads();
}

__global__ __launch_bounds__(256) void cstr_rollout_kernel(
    const float* __restrict__ useq, const float* __restrict__ xGz0,
    const float* __restrict__ W1g, const float* __restrict__ b1g,
    const float* __restrict__ W2g, const float* __restrict__ b2g,
    const float* __restrict__ W3g, const float* __restrict__ b3g,
    float* __restrict__ out) {
  extern __shared__ __align__(16) char smem[];
  v2f*   W1p  = (v2f*)(smem + OFF_W1P);
  v2f*   W2p  = (v2f*)(smem + OFF_W2P);
  v2f*   W3p  = (v2f*)(smem + OFF_W3P);
  float* sInp = (float*)(smem + OFF_INP);
  float* sH1  = (float*)(smem + OFF_H1);
  float* sH2  = (float*)(smem + OFF_H2);
  float* sNN  = (float*)(smem + OFF_NN);
  float* sXG  = (float*)(smem + OFF_XG);
  float* sXs  = (float*)(smem + OFF_XS);
  float* sZ   = (float*)(smem + OFF_Z);   // stride 52, 50 used
  float* sK   = (float*)(smem + OFF_K);
  float* sU   = (float*)(smem + OFF_U);
  float* sB1  = (float*)(smem + OFF_B1);
  float* sB2  = (float*)(smem + OFF_B2);
  float* sB3  = (float*)(smem + OFF_B3);

  const int tid  = threadIdx.x;
  const int lane = tid & 31;
  const int wave = tid >> 5;
  const int bt0  = blockIdx.x * BT;

  // ---- load + swizzle weights into LDS (once) ----
  for (int i = tid; i < 32 * 128; i += 256) {        // W1: [60x128] -> pairs, K pad 64
    int kk = i >> 7, n = i & 127;
    float lo = (2 * kk     < 60) ? W1g[(2 * kk)     * 128 + n] : 0.0f;
    float hi = (2 * kk + 1 < 60) ? W1g[(2 * kk + 1) * 128 + n] : 0.0f;
    W1p[kk * LDW + n] = (v2f){lo, hi};
  }
  for (int i = tid; i < 64 * 128; i += 256) {        // W2: [128x128] -> pairs
    int kk = i >> 7, n = i & 127;
    W2p[kk * LDW + n] = (v2f){W2g[(2 * kk) * 128 + n], W2g[(2 * kk + 1) * 128 + n]};
  }
  for (int i = tid; i < 64 * 16; i += 256) {         // W3: [128x8] -> pairs, N pad 16
    int kk = i >> 4, n = i & 15;
    v2f p = (v2f){0.0f, 0.0f};
    if (n < NXc) p = (v2f){W3g[(2 * kk) * NXc + n], W3g[(2 * kk + 1) * NXc + n]};
    W3p[kk * LDW3 + n] = p;
  }
  if (tid < 128) { sB1[tid] = b1g[tid]; sB2[tid] = b2g[tid]; }
  if (tid < NXc) sB3[tid] = b3g[tid];
  // ---- initial state ----
  for (int i = tid; i < BT * 58; i += 256) {
    int r = i / 58, c = i - r * 58;
    float v = xGz0[(size_t)(bt0 + r) * 58 + c];
    if (c < NXc) sXG[r * NXc + c] = v;
    else         sZ[r * 52 + (c - NXc)] = v;
  }
  __syncthreads();

  for (int t = 0; t < T_STEPS; ++t) {
    // u for this step (raw scaled values, as fed to fnn)
    if (tid < BT * NUc) {
      int r = tid >> 1, c = tid & 1;
      sU[tid] = useq[(size_t)(bt0 + r) * T_STEPS * NUc + (size_t)t * NUc + c];
    }
    __syncthreads();

    // ---- stage 1: inp = [xG, z, u, 0] ----
    for (int i = tid; i < BT * DINP; i += 256) {
      int r = i >> 6, c = i & 63;
      float v;
      if      (c < 8)  v = sXG[r * NXc + c];
      else if (c < 58) v = sZ[r * 52 + (c - 8)];
      else if (c < 60) v = sU[r * 2 + (c - 58)];
      else             v = 0.0f;
      sInp[i] = v;
    }
    __syncthreads();
    mlp_and_physics(tid, wave, lane, 0, sInp, sH1, sH2, sNN, W1p, W2p, W3p,
                    sB1, sB2, sB3, sXG, sU, sK);

    // ---- stage 2: x2 = xG + 0.5*dt*k1 ; z_half = [interp(yp,xG), upseq] ----
    for (int i = tid; i < BT * DINP; i += 256) {
      int r = i >> 6, c = i & 63;
      float v;
      if (c < 8) {
        v = sXG[r * NXc + c] + 0.5f * DELTA_F * sK[0 * 128 + r * NXc + c];
        sXs[r * NXc + c] = v;
      } else if (c < 48) {
        int idx = c - 8, p = idx >> 3, q = idx & 7;
        float y0 = sZ[r * 52 + p * 8 + q];
        float y1 = (p < 4) ? sZ[r * 52 + (p + 1) * 8 + q] : sXG[r * NXc + q];
        v = 0.5f * (y0 + y1);
      } else if (c < 58) v = sZ[r * 52 + 40 + (c - 48)];
      else if (c < 60)   v = sU[r * 2 + (c - 58)];
      else               v = 0.0f;
      sInp[i] = v;
    }
    __syncthreads();
    mlp_and_physics(tid, wave, lane, 1, sInp, sH1, sH2, sNN, W1p, W2p, W3p,
                    sB1, sB2, sB3, sXs, sU, sK);

    // ---- stage 3: only x3 changes (z_half reused in sInp cols 8..63) ----
    if (tid < BT * NXc) {
      int r = tid >> 3, j = tid & 7;
      float v = sXG[r * NXc + j] + 0.5f * DELTA_F * sK[1 * 128 + r * NXc + j];
      sXs[r * NXc + j] = v;
      sInp[r * DINP + j] = v;
    }
    __syncthreads();
    mlp_and_physics(tid, wave, lane, 2, sInp, sH1, sH2, sNN, W1p, W2p, W3p,
                    sB1, sB2, sB3, sXs, sU, sK);

    // ---- stage 4: x4 = xG + dt*k3 ; z_full = [z[8:40], xG, upseq] ----
    for (int i = tid; i < BT * 48; i += 256) {
      int r = i / 48, c = i - r * 48;
      float v;
      if (c < 8) {
        v = sXG[r * NXc + c] + DELTA_F * sK[2 * 128 + r * NXc + c];
        sXs[r * NXc + c] = v;
      } else {
        int idx = c - 8;
        v = (idx < 32) ? sZ[r * 52 + 8 + idx] : sXG[r * NXc + (idx - 32)];
      }
      sInp[r * DINP + c] = v;
    }
    __syncthreads();
    mlp_and_physics(tid, wave, lane, 3, sInp, sH1, sH2, sNN, W1p, W2p, W3p,
                    sB1, sB2, sB3, sXs, sU, sK);

    // ---- epilogue: y_t = xG ; xG += dt/6*(k1+2k2+2k3+k4) ; shift z ----
    float xgnew = 0.0f;
    if (tid < BT * NXc) {
      int r = tid >> 3, j = tid & 7;
      float xg = sXG[r * NXc + j];
      out[(size_t)(bt0 + r) * T_STEPS * NXc + (size_t)t * NXc + j] = xg;
      xgnew = xg + (DELTA_F / 6.0f) *
              (sK[0 * 128 + tid] + 2.0f * sK[1 * 128 + tid] +
               2.0f * sK[2 * 128 + tid] + sK[3 * 128 + tid]);
    }
    float regz[4];
#pragma unroll
    for (int ii = 0; ii < 4; ++ii) {
      int i = tid + ii * 256;
      if (i < BT * NZc) {
        int r = i / NZc, c = i - r * NZc;
        float v;
        if      (c < 32) v = sZ[r * 52 + 8 + c];        // ypseq shift
        else if (c < 40) v = sXG[r * NXc + (c - 32)];   // append old xG
        else if (c < 48) v = sZ[r * 52 + (c + 2)];      // upseq shift
        else             v = sU[r * 2 + (c - 48)];      // append u
        regz[ii] = v;
      }
    }
    __syncthreads();
#pragma unroll
    for (int ii = 0; ii < 4; ++ii) {
      int i = tid + ii * 256;
      if (i < BT * NZc) {
        int r = i / NZc, c = i - r * NZc;
        sZ[r * 52 + c] = regz[ii];
      }
    }
    if (tid < BT * NXc) sXG[tid] = xgnew;
    __syncthreads();
  }
}

extern "C" void kernel_launch(void* const* d_in, const int* in_sizes, int n_in,
                              void* d_out, int out_size, void* d_ws, size_t ws_size,
                              hipStream_t stream) {
  (void)in_sizes; (void)n_in; (void)out_size; (void)d_ws; (void)ws_size;
  const float* useq = (const float*)d_in[0];
  const float* xGz0 = (const float*)d_in[1];
  const float* W1   = (const float*)d_in[2];
  const float* b1   = (const float*)d_in[3];
  const float* W2   = (const float*)d_in[4];
  const float* b2   = (const float*)d_in[5];
  const float* W3   = (const float*)d_in[6];
  const float* b3   = (const float*)d_in[7];
  float* out = (float*)d_out;
  cstr_rollout_kernel<<<dim3(B_TOT / BT), dim3(256), SMEM_BYTES, stream>>>(
      useq, xGz0, W1, b1, W2, b2, W3, b3, out);
}